// GCNConv_52553219833884
// MI455X (gfx1250) — compile-verified
//
#include <hip/hip_runtime.h>

typedef __attribute__((ext_vector_type(2))) float v2f;
typedef __attribute__((ext_vector_type(8))) float v8f;

#define N_NODES 10000
#define C_IN    128
#define C_OUT   256
#define N_EDGES 640000

// ---------------------------------------------------------------------------
// Kernel 1: zero the aggregation buffer (d_ws). float4 stores, 320k vec4s.
// ---------------------------------------------------------------------------
__global__ void zero_agg(float4* __restrict__ agg4) {
    int i = blockIdx.x * blockDim.x + threadIdx.x;
    if (i < (N_NODES * C_IN) / 4) {
        agg4[i] = make_float4(0.f, 0.f, 0.f, 0.f);
    }
}

// ---------------------------------------------------------------------------
// Kernel 2: edge-parallel scatter-add.  One wave32 per edge; each lane moves
// 4 consecutive floats of the 128-wide feature row via float4 load + 4
// global_atomic_add_f32.  features (5MB) and agg (5MB) are L2-resident.
// ---------------------------------------------------------------------------
__global__ void scatter_add(const float* __restrict__ feat,
                            const long long* __restrict__ src,
                            const long long* __restrict__ dst,
                            float* __restrict__ agg) {
    long long gid = (long long)blockIdx.x * blockDim.x + threadIdx.x;
    int e    = (int)(gid >> 5);   // edge index (32 lanes per edge)
    int lane = (int)(gid & 31);
    int s = (int)src[e];
    int d = (int)dst[e];
    float4 v = ((const float4*)(feat + (size_t)s * C_IN))[lane];
    float* o = agg + (size_t)d * C_IN + lane * 4;
    atomicAdd(o + 0, v.x);
    atomicAdd(o + 1, v.y);
    atomicAdd(o + 2, v.z);
    atomicAdd(o + 3, v.w);
}

// ---------------------------------------------------------------------------
// Kernel 3: f32 GEMM  out[10000,256] = agg[10000,128] @ W[128,256] + b
// using V_WMMA_F32_16X16X4_F32.  One wave per 16x16 output tile.
// 625 row-tiles x 16 col-tiles = 10000 tiles; 8 waves/block -> 1250 blocks
// (exact, no tail -> EXEC stays all-ones for WMMA).
//
// Fragment layouts per CDNA5 ISA 7.12.2:
//   A 16x4 (MxK) f32, 2 VGPRs: lanes 0-15 M=lane (VGPR0:K=0, VGPR1:K=1),
//                              lanes 16-31 M=lane-16 (VGPR0:K=2, VGPR1:K=3)
//   B 4x16 (KxN) f32, 2 VGPRs: N = lane&15, K = vgpr + 2*(lane>=16)
//   C/D 16x16 f32, 8 VGPRs:    N = lane&15, M = vgpr + 8*(lane>=16)
// ---------------------------------------------------------------------------
__global__ void gemm_wmma(const float* __restrict__ A,     // [N_NODES, C_IN]
                          const float* __restrict__ W,     // [C_IN, C_OUT]
                          const float* __restrict__ bias,  // [C_OUT]
                          float* __restrict__ out) {       // [N_NODES, C_OUT]
    int wave = threadIdx.x >> 5;                 // 0..7
    int lane = threadIdx.x & 31;
    int tile = blockIdx.x * 8 + wave;            // 0..9999
    int tileN = tile & 15;                       // 16 col tiles
    int tileM = tile >> 4;                       // 625 row tiles

    int half = lane >> 4;                        // 0 or 1
    int l15  = lane & 15;

    int rowA = tileM * 16 + l15;                 // A-fragment row (M)
    int colB = tileN * 16 + l15;                 // B/D-fragment col (N)
    const float* Arow = A + (size_t)rowA * C_IN;

    v8f c = {};
    #pragma unroll
    for (int k0 = 0; k0 < C_IN; k0 += 4) {
        int kk = k0 + 2 * half;                  // K for fragment VGPR0
        v2f a, b;
        a.x = Arow[kk];
        a.y = Arow[kk + 1];
        b.x = W[(size_t)kk       * C_OUT + colB];
        b.y = W[(size_t)(kk + 1) * C_OUT + colB];
        // 8 args: (neg_a, A, neg_b, B, c_mod, C, reuse_a, reuse_b)
        c = __builtin_amdgcn_wmma_f32_16x16x4_f32(
                /*neg_a=*/false, a, /*neg_b=*/false, b,
                /*c_mod=*/(short)0, c, /*reuse_a=*/false, /*reuse_b=*/false);
    }

    float bb = bias[colB];
    #pragma unroll
    for (int v = 0; v < 8; ++v) {
        int row = tileM * 16 + v + 8 * half;     // D-matrix M per layout
        out[(size_t)row * C_OUT + colB] = c[v] + bb;
    }
}

// ---------------------------------------------------------------------------
extern "C" void kernel_launch(void* const* d_in, const int* in_sizes, int n_in,
                              void* d_out, int out_size, void* d_ws, size_t ws_size,
                              hipStream_t stream) {
    const float*     feat = (const float*)d_in[0];       // [10000,128] f32
    const long long* edge = (const long long*)d_in[1];   // [2,640000] i64
    const float*     W    = (const float*)d_in[2];       // [128,256] f32
    const float*     bias = (const float*)d_in[3];       // [256] f32
    float*           out  = (float*)d_out;               // [10000,256] f32
    float*           agg  = (float*)d_ws;                // [10000,128] scratch

    // 1) zero accumulator (harness poisons d_ws, never re-zeroes)
    {
        int n4 = (N_NODES * C_IN) / 4;                   // 320000
        zero_agg<<<(n4 + 255) / 256, 256, 0, stream>>>((float4*)agg);
    }
    // 2) scatter-add: 640k edges * 32 lanes = 20.48M threads
    {
        long long threads = (long long)N_EDGES * 32;
        int blocks = (int)(threads / 256);               // 80000, exact
        scatter_add<<<blocks, 256, 0, stream>>>(feat, edge, edge + N_EDGES, agg);
    }
    // 3) WMMA GEMM: 10000 tiles / 8 waves per block = 1250 blocks, exact
    gemm_wmma<<<1250, 256, 0, stream>>>(agg, W, bias, out);
}